// DCE_1133871366378
// MI455X (gfx1250) — compile-verified
//
#include <hip/hip_runtime.h>
#include <math.h>

// DCE forward model for MI455X (gfx1250, wave32).
//   stage 1: build shifted-DFT matrix G (re/im), G = (-1)^(a+b) * W^(ab) / sqrt(N)
//   stage 2: per-pixel 2-compartment Euler ODE + SPGR signal model -> sig[50][320][320]
//   stage 3: Y = G * sig      (f32 WMMA 16x16x4, 2 accumulators share B-frag loads)
//   stage 4: R = Y * G        (complex, 4 WMMAs/step, 3 accumulators) -> d_out (complex64)
// Whole working set (~83 MB) lives in the 192 MB L2; the DFT-as-matmul keeps full fp32
// precision, which is why we use the f32 16x16x4 WMMA shape instead of bf16/fp8.

typedef float v2f __attribute__((ext_vector_type(2)));
typedef float v8f __attribute__((ext_vector_type(8)));

#define NPIX   (320 * 320)
#define NFR    50
#define TSAMP  295
#define LDN    320

__device__ __forceinline__ v8f wmma_f32_k4(v2f a, v2f b, v8f c) {
  // D = A(16x4,f32) * B(4x16,f32) + C(16x16,f32); 8 args like the f16/bf16 forms.
  return __builtin_amdgcn_wmma_f32_16x16x4_f32(
      /*neg_a=*/false, a, /*neg_b=*/false, b,
      /*c_mod=*/(short)0, c, /*reuse_a=*/false, /*reuse_b=*/false);
}

// ---------------- stage 1: shifted DFT matrix -------------------------------
__global__ void build_G(float* __restrict__ Gr, float* __restrict__ Gi) {
  int idx = blockIdx.x * blockDim.x + threadIdx.x;
  if (idx >= NPIX) return;
  int a = idx / LDN, b = idx % LDN;
  // exp(-2*pi*i*(a+160)(b+160)/320) = (-1)^(a+b) * exp(-2*pi*i*(a*b mod 320)/320)
  int m = (a * b) % LDN;                      // exact integer argument reduction
  float ang = -6.283185307179586f * (float)m / 320.0f;
  float s, c;
  sincosf(ang, &s, &c);
  float sgn = ((a + b) & 1) ? -1.0f : 1.0f;
  const float scale = 0.05590169943749474f;   // 1/sqrt(320), ortho norm
  Gr[idx] = sgn * c * scale;
  Gi[idx] = sgn * s * scale;
}

// ---------------- stage 2: ODE + SPGR signal --------------------------------
__global__ void ode_signal(const float* __restrict__ x,
                           const float* __restrict__ aif,
                           float* __restrict__ sig) {
  __shared__ float s_aif[TSAMP];
  for (int i = threadIdx.x; i < TSAMP; i += blockDim.x) s_aif[i] = aif[i];
  __syncthreads();

  int pix = blockIdx.x * blockDim.x + threadIdx.x;
  if (pix >= NPIX) return;

  float ve = x[pix];
  float vp = x[NPIX + pix];
  float fp = x[2 * NPIX + pix];
  float PS = x[3 * NPIX + pix];
  float inv_vp = 1.0f / vp;   // dt = 1
  float inv_ve = 1.0f / ve;

  const float COS_FA = 0.9659258262890683f;   // cos(15 deg)
  const float E1     = 0.9940179485512251f;   // exp(-TR*R1), TR=0.006, R1=1
  const float M0T    = (1.0f - COS_FA * E1) / (1.0f - E1);      // M0_TRANS (SIG_BASELINE=1)
  const float MSTDY  = M0T * (1.0f - E1) / (1.0f - E1 * COS_FA);

  float cp = 0.0f, ce = 0.0f;
  { // frame 0: conc == 0
    float e = E1;
    sig[pix] = M0T * (1.0f - e) / (1.0f - e * COS_FA) + 1.0f - MSTDY;
  }
  int t = 0;
  for (int fr = 1; fr < NFR; ++fr) {
#pragma unroll
    for (int s6 = 0; s6 < 6; ++s6) {
      float a    = s_aif[t];
      float d_cp = fp * a + PS * ce - (fp + PS) * cp;
      float d_ce = PS * (cp - ce);
      cp += d_cp * inv_vp;
      ce += d_ce * inv_ve;
      ++t;
    }
    float conc = vp * cp + ve * ce;
    if (!(conc == conc)) conc = 0.0f;         // NaN -> 0 (matches jnp.where)
    float e = expf(-0.006f * (1.0f + 4.3f * conc));
    sig[fr * NPIX + pix] = M0T * (1.0f - e) / (1.0f - e * COS_FA) + 1.0f - MSTDY;
  }
}

// ---------------- WMMA fragment helpers (ISA 32-bit A/B layouts) ------------
__device__ __forceinline__ v2f load_a(const float* __restrict__ A, int row, int kk) {
  // A 16x4: lanes 0-15 hold K={k0,k0+1}, lanes 16-31 K={k0+2,k0+3}; contiguous pair.
  return *(const v2f*)(A + row * LDN + kk);
}
__device__ __forceinline__ v2f load_b(const float* __restrict__ B, int col, int kk) {
  // B 4x16: lane half selects K base, VGPR index adds +1; N striped over 16 lanes.
  v2f r;
  r.x = B[kk * LDN + col];
  r.y = B[(kk + 1) * LDN + col];
  return r;
}

// ---------------- stage 3: Y = G * sig (one 16x16 tile per wave) ------------
__global__ void mm_stageA(const float* __restrict__ Gr, const float* __restrict__ Gi,
                          const float* __restrict__ sig,
                          float* __restrict__ Yr, float* __restrict__ Yi) {
  int wave = (blockIdx.x * blockDim.x + threadIdx.x) >> 5;
  int lane = threadIdx.x & 31;
  if (wave >= NFR * 400) return;
  int f    = wave / 400;
  int tile = wave % 400;
  int tm = (tile / 20) * 16;
  int tn = (tile % 20) * 16;

  const float* B = sig + f * NPIX;
  int row   = tm + (lane & 15);
  int col   = tn + (lane & 15);
  int khalf = (lane >> 4) << 1;

  v8f accR = {}, accI = {};
#pragma unroll 4
  for (int k0 = 0; k0 < LDN; k0 += 4) {
    int kk = k0 + khalf;
    v2f ar = load_a(Gr, row, kk);
    v2f ai = load_a(Gi, row, kk);
    v2f bf = load_b(B, col, kk);
    accR = wmma_f32_k4(ar, bf, accR);
    accI = wmma_f32_k4(ai, bf, accI);
  }

  float* yr = Yr + f * NPIX;
  float* yi = Yi + f * NPIX;
  int rbase = tm + ((lane >> 4) << 3);        // C/D: lane hi-half -> M+8
  int c     = tn + (lane & 15);
#pragma unroll
  for (int v = 0; v < 8; ++v) {
    yr[(rbase + v) * LDN + c] = accR[v];
    yi[(rbase + v) * LDN + c] = accI[v];
  }
}

// ---------------- stage 4: R = Y * G -> interleaved complex64 ---------------
__global__ void mm_stageB(const float* __restrict__ Gr, const float* __restrict__ Gi,
                          const float* __restrict__ Yr, const float* __restrict__ Yi,
                          float* __restrict__ out) {
  int wave = (blockIdx.x * blockDim.x + threadIdx.x) >> 5;
  int lane = threadIdx.x & 31;
  if (wave >= NFR * 400) return;
  int f    = wave / 400;
  int tile = wave % 400;
  int tm = (tile / 20) * 16;
  int tn = (tile % 20) * 16;

  const float* Ar = Yr + f * NPIX;
  const float* Ai = Yi + f * NPIX;
  int row   = tm + (lane & 15);
  int col   = tn + (lane & 15);
  int khalf = (lane >> 4) << 1;

  // f32 WMMA has no A/B negate (NEG is CNeg only) -> keep Yi*Gi in its own
  // accumulator and subtract once at the end.
  v8f accRR = {}, accII = {}, accRI = {};
#pragma unroll 4
  for (int k0 = 0; k0 < LDN; k0 += 4) {
    int kk = k0 + khalf;
    v2f ar = load_a(Ar, row, kk);
    v2f ai = load_a(Ai, row, kk);
    v2f br = load_b(Gr, col, kk);
    v2f bi = load_b(Gi, col, kk);
    accRR = wmma_f32_k4(ar, br, accRR);
    accII = wmma_f32_k4(ai, bi, accII);
    accRI = wmma_f32_k4(ar, bi, accRI);
    accRI = wmma_f32_k4(ai, br, accRI);
  }

  float2* oc = (float2*)out + f * NPIX;       // complex64 interleaved
  int rbase = tm + ((lane >> 4) << 3);
  int c     = tn + (lane & 15);
#pragma unroll
  for (int v = 0; v < 8; ++v) {
    float re = accRR[v] - accII[v];
    float im = accRI[v];
    oc[(rbase + v) * LDN + c] = make_float2(re, im);
  }
}

// ---------------- launcher ---------------------------------------------------
extern "C" void kernel_launch(void* const* d_in, const int* in_sizes, int n_in,
                              void* d_out, int out_size, void* d_ws, size_t ws_size,
                              hipStream_t stream) {
  const float* x   = (const float*)d_in[0];   // [4,1,320,320] f32
  const float* aif = (const float*)d_in[1];   // [295] f32
  (void)in_sizes; (void)n_in; (void)out_size; (void)ws_size;

  float* ws = (float*)d_ws;
  float* Gr = ws;                             //   0.41 MB
  float* Gi = ws + NPIX;                      //   0.41 MB
  float* Yr = ws + 2 * NPIX;                  //  20.48 MB
  float* Yi = Yr + NFR * NPIX;                //  20.48 MB
  float* sigbuf = (float*)d_out;              // stage sig in d_out front half;
                                              // dead before stage 4 overwrites it.

  build_G  <<<(NPIX + 255) / 256, 256, 0, stream>>>(Gr, Gi);
  ode_signal<<<(NPIX + 255) / 256, 256, 0, stream>>>(x, aif, sigbuf);

  int waves  = NFR * 400;                     // 20000 tiles, 4 waves/block
  int blocks = waves / 4;
  mm_stageA<<<blocks, 128, 0, stream>>>(Gr, Gi, sigbuf, Yr, Yi);
  mm_stageB<<<blocks, 128, 0, stream>>>(Gr, Gi, Yr, Yi, (float*)d_out);
}